// IsingTransformerLayer_15135464751131
// MI455X (gfx1250) — compile-verified
//
#include <hip/hip_runtime.h>
#include <hip/hip_bf16.h>

// ---------------------------------------------------------------------------
// Ising transformer layer for MI455X (gfx1250), wave32 + WMMA f32 16x16x4,
// async global->LDS staging (ASYNCcnt double buffering).
//
// 168 inversions of V = diag(t) - J (1024x1024) are replaced by
//   X1 = D^-1 + D^-1 J D^-1        (elementwise; residual ~ (1/256)^2)
//   X  <- X (2I - V X)  (twice)    (pure GEMM; residual -> ~1e-10)
// ---------------------------------------------------------------------------

typedef __attribute__((ext_vector_type(2))) float v2f;
typedef __attribute__((ext_vector_type(8))) float v8f;

#define NTOK 1024
#define DIMF 512
#define NHEAD 8
#define DHEAD 64

// async copy of 16 bytes, global -> LDS, tracked by ASYNCcnt
#define ASYNC_LOAD_B128(ldsoff, gptr)                                        \
    asm volatile("global_load_async_to_lds_b128 %0, %1, off"                 \
                 :: "v"(ldsoff), "v"(gptr) : "memory")

// ---------------------------------------------------------------------------
// WMMA GEMM:  acc = A(MxK) * op(B)(KxN).  Block tile 128x64, 8 waves, each
// wave owns a 16x64 slab (4 accumulators).  K staged through LDS in tiles of
// 16 with async double buffering (3 async b128 ops per thread per tile).
// mode 0: C = alpha * acc
// mode 1: C = t[i]*aux[i,j] - acc      (Y = V*X, V = diag(t)-J, A=J)
// mode 2: C = 2*aux[i,j]   - acc       (Newton-Schulz: X' = 2X - X*Y, A=X)
// ---------------------------------------------------------------------------
template <int TRANSB>
__global__ __launch_bounds__(256) void gemm_f32_wmma(
    const float* __restrict__ A, int lda,
    const float* __restrict__ B, int ldb,
    float* __restrict__ C, int ldc,
    const float* __restrict__ aux, int ldaux,
    const float* __restrict__ tv,
    int K, int mode, float alpha)
{
    // A: 128 rows x 16 k, row stride 20 floats (80B: 16B aligned, 20i%64
    //    distinct for i<16 -> conflict-free fragment reads).
    // B: KN layout (stride 64) or NK layout (stride 20) depending on TRANSB.
    __shared__ float sA[2][128 * 20];
    __shared__ float sB[2][64 * 20];

    const int tid  = threadIdx.x;
    const int lane = tid & 31;
    const int wave = tid >> 5;       // 0..7 -> M sub-tile
    const int half = lane >> 4;      // K-half selector of f32 WMMA layout
    const int l16  = lane & 15;
    const int m0   = blockIdx.y * 128;
    const int n0   = blockIdx.x * 64;

    // per-thread async-copy coordinates (256 threads move the whole tile)
    const int a_m  = tid >> 2;       // 0..63 (plus +64 for second chunk)
    const int a_kq = tid & 3;        // which 4-float group of the 16-wide k
    const int b_k  = tid >> 4;       // KN: k row 0..15
    const int b_nq = tid & 15;       // KN: 4-float group of 64-wide n
    const int bt_n = tid >> 2;       // NK: n row 0..63
    const int bt_kq = tid & 3;       // NK: 4-float group of 16-wide k

    auto issue = [&](int k0, int buf) {
        unsigned baseA = (unsigned)(size_t)&sA[buf][0];
        unsigned baseB = (unsigned)(size_t)&sB[buf][0];
        const float* ga0 = A + (size_t)(m0 + a_m) * lda + k0 + a_kq * 4;
        const float* ga1 = A + (size_t)(m0 + 64 + a_m) * lda + k0 + a_kq * 4;
        ASYNC_LOAD_B128(baseA + a_m * 80 + a_kq * 16, ga0);
        ASYNC_LOAD_B128(baseA + (64 + a_m) * 80 + a_kq * 16, ga1);
        if (TRANSB) {
            const float* gb = B + (size_t)(n0 + bt_n) * ldb + k0 + bt_kq * 4;
            ASYNC_LOAD_B128(baseB + bt_n * 80 + bt_kq * 16, gb);
        } else {
            const float* gb = B + (size_t)(k0 + b_k) * ldb + n0 + b_nq * 4;
            ASYNC_LOAD_B128(baseB + b_k * 256 + b_nq * 16, gb);
        }
    };

    v8f acc[4] = {};
    const int nt = K >> 4;

    issue(0, 0);
    for (int kt = 0; kt < nt; ++kt) {
        const int buf = kt & 1;
        if (kt + 1 < nt) {
            issue((kt + 1) << 4, buf ^ 1);
            // 3 newest async ops are tile kt+1; everything older has landed
            asm volatile("s_wait_asynccnt 0x3" ::: "memory");
        } else {
            asm volatile("s_wait_asynccnt 0x0" ::: "memory");
        }
        __syncthreads();

        const float* Ab = &sA[buf][0];
        const float* Bb = &sB[buf][0];
        #pragma unroll
        for (int kk = 0; kk < 16; kk += 4) {
            v2f a;
            a.x = Ab[(wave * 16 + l16) * 20 + kk + 2 * half + 0];
            a.y = Ab[(wave * 16 + l16) * 20 + kk + 2 * half + 1];
            #pragma unroll
            for (int j = 0; j < 4; ++j) {
                v2f b;
                if (TRANSB) {
                    b.x = Bb[(j * 16 + l16) * 20 + kk + 2 * half + 0];
                    b.y = Bb[(j * 16 + l16) * 20 + kk + 2 * half + 1];
                } else {
                    b.x = Bb[(kk + 2 * half + 0) * 64 + j * 16 + l16];
                    b.y = Bb[(kk + 2 * half + 1) * 64 + j * 16 + l16];
                }
                acc[j] = __builtin_amdgcn_wmma_f32_16x16x4_f32(
                    false, a, false, b, (short)0, acc[j], false, false);
            }
        }
        __syncthreads();
    }

    // epilogue with fused Newton-Schulz / V-apply forms
    #pragma unroll
    for (int j = 0; j < 4; ++j) {
        #pragma unroll
        for (int r = 0; r < 8; ++r) {
            int gi = m0 + wave * 16 + 8 * half + r;
            int gj = n0 + j * 16 + l16;
            float v = acc[j][r];
            if (mode == 0) {
                v *= alpha;
            } else if (mode == 1) {
                v = tv[gi] * aux[(size_t)gi * ldaux + gj] - v;
            } else {
                v = 2.0f * aux[(size_t)gi * ldaux + gj] - v;
            }
            C[(size_t)gi * ldc + gj] = v;
        }
    }
}

// ---------------------------------------------------------------------------
// LayerNorm row kernel: h = ((x-mu)/sqrt(var+eps)*w + b) / sqrt(DHEAD)
// ---------------------------------------------------------------------------
__global__ __launch_bounds__(256) void layernorm_kernel(
    const float* __restrict__ x, const float* __restrict__ w,
    const float* __restrict__ b, float* __restrict__ h, int dim)
{
    __shared__ float r1[256], r2[256];
    const int row = blockIdx.x;
    float s = 0.f, s2 = 0.f;
    for (int j = threadIdx.x; j < dim; j += 256) {
        float v = x[(size_t)row * dim + j];
        s += v; s2 += v * v;
    }
    r1[threadIdx.x] = s; r2[threadIdx.x] = s2;
    __syncthreads();
    for (int st = 128; st > 0; st >>= 1) {
        if (threadIdx.x < st) {
            r1[threadIdx.x] += r1[threadIdx.x + st];
            r2[threadIdx.x] += r2[threadIdx.x + st];
        }
        __syncthreads();
    }
    float mean = r1[0] / dim;
    float var  = r2[0] / dim - mean * mean;
    float rstd = rsqrtf(var + 1e-5f);
    for (int j = threadIdx.x; j < dim; j += 256) {
        float v = x[(size_t)row * dim + j];
        h[(size_t)row * dim + j] = ((v - mean) * rstd * w[j] + b[j]) * 0.125f;
    }
}

// ---------------------------------------------------------------------------
// Rotary embedding (rotate_every_two), in-place; one thread per pair.
// ---------------------------------------------------------------------------
__global__ __launch_bounds__(256) void rope_kernel(
    float* __restrict__ buf, const float* __restrict__ sinb,
    const float* __restrict__ cosb, int n, int width)
{
    int idx = blockIdx.x * blockDim.x + threadIdx.x;
    int total = n * (width >> 1);
    if (idx >= total) return;
    int row = idx / (width >> 1);
    int c   = (idx % (width >> 1)) * 2;
    int dd  = c & (DHEAD - 1);
    size_t base = (size_t)row * width + c;
    float v0 = buf[base], v1 = buf[base + 1];
    float sn0 = sinb[row * DHEAD + dd],     cs0 = cosb[row * DHEAD + dd];
    float sn1 = sinb[row * DHEAD + dd + 1], cs1 = cosb[row * DHEAD + dd + 1];
    buf[base]     = v0 * cs0 - v1 * sn0;
    buf[base + 1] = v1 * cs1 + v0 * sn1;
}

// ---------------------------------------------------------------------------
// Causal softmax row kernel (in place): J = softmax(masked S) * outscale.
// ---------------------------------------------------------------------------
__global__ __launch_bounds__(256) void softmax_causal(
    float* __restrict__ S, int n, float outscale)
{
    __shared__ float red[256];
    const int row = blockIdx.x;
    const int valid = row + 1;
    float m = -3.0e38f;
    for (int j = threadIdx.x; j < valid; j += 256)
        m = fmaxf(m, S[(size_t)row * n + j]);
    red[threadIdx.x] = m;
    __syncthreads();
    for (int st = 128; st > 0; st >>= 1) {
        if (threadIdx.x < st)
            red[threadIdx.x] = fmaxf(red[threadIdx.x], red[threadIdx.x + st]);
        __syncthreads();
    }
    m = red[0];
    __syncthreads();
    float sum = 0.f;
    for (int j = threadIdx.x; j < valid; j += 256)
        sum += __expf(S[(size_t)row * n + j] - m);
    red[threadIdx.x] = sum;
    __syncthreads();
    for (int st = 128; st > 0; st >>= 1) {
        if (threadIdx.x < st) red[threadIdx.x] += red[threadIdx.x + st];
        __syncthreads();
    }
    float inv = outscale / red[0];
    for (int j = threadIdx.x; j < n; j += 256) {
        size_t p = (size_t)row * n + j;
        S[p] = (j < valid) ? __expf(S[p] - m) * inv : 0.0f;
    }
}

// X1 = D^-1 + D^-1 J D^-1   (first Newton-Schulz step, closed form)
__global__ __launch_bounds__(256) void build_x1(
    float* __restrict__ X, const float* __restrict__ J,
    const float* __restrict__ t, int n)
{
    int idx = blockIdx.x * blockDim.x + threadIdx.x;
    int i = idx / n, j = idx % n;
    float ti = t[i], tj = t[j];
    float v = J[idx] / (ti * tj);
    if (i == j) v += 1.0f / ti;
    X[idx] = v;
}

// t <- 0.5 t + 0.5 * (0.5 diag(Vinv) + 0.25 * rowsumsq(Vinv h))
__global__ __launch_bounds__(256) void update_t(
    float* __restrict__ t, const float* __restrict__ X,
    const float* __restrict__ W, int n)
{
    int i = blockIdx.x * blockDim.x + threadIdx.x;
    if (i >= n) return;
    float d = X[(size_t)i * n + i];
    float s = 0.f;
    #pragma unroll
    for (int f = 0; f < DHEAD; ++f) {
        float w = W[(size_t)i * DHEAD + f];
        s += w * w;
    }
    float g = 0.5f * d + 0.25f * s;   // beta = 1
    t[i] = 0.5f * t[i] + 0.5f * g;    // MIX = 0.5
}

__global__ __launch_bounds__(256) void set_ones(float* __restrict__ t, int n)
{
    int i = blockIdx.x * blockDim.x + threadIdx.x;
    if (i < n) t[i] = 1.0f;
}

// Y = -0.25 * (X + X^T)
__global__ __launch_bounds__(256) void symneg(
    float* __restrict__ Y, const float* __restrict__ X, int n)
{
    int idx = blockIdx.x * blockDim.x + threadIdx.x;
    int i = idx / n, j = idx % n;
    Y[idx] = -0.25f * (X[idx] + X[(size_t)j * n + i]);
}

// ---------------------------------------------------------------------------
extern "C" void kernel_launch(void* const* d_in, const int* in_sizes, int n_in,
                              void* d_out, int out_size, void* d_ws, size_t ws_size,
                              hipStream_t stream)
{
    const float* x   = (const float*)d_in[0];
    const float* wq  = (const float*)d_in[1];
    const float* wk  = (const float*)d_in[2];
    const float* lnw = (const float*)d_in[3];
    const float* lnb = (const float*)d_in[4];
    const float* sn  = (const float*)d_in[5];
    const float* cs  = (const float*)d_in[6];
    float* out = (float*)d_out;

    const int n = NTOK, dim = DIMF;

    // workspace partition (~23.3 MB of f32)
    float* h  = (float*)d_ws;            // n*dim
    float* q  = h + (size_t)n * dim;     // n*dim
    float* k  = q + (size_t)n * dim;     // n*dim
    float* t  = k + (size_t)n * dim;     // n (padded to 1024)
    float* J  = t + 1024;                // n*n
    float* X  = J + (size_t)n * n;       // n*n
    float* X2 = X + (size_t)n * n;       // n*n
    float* Y  = X2 + (size_t)n * n;      // n*n
    float* W  = Y + (size_t)n * n;       // n*DHEAD

    auto gemm = [&](const float* A, int lda, const float* B, int ldb,
                    float* C, int ldc, int M, int N, int K,
                    int transB, int mode, float alpha,
                    const float* aux, int ldaux, const float* tv) {
        dim3 grid(N / 64, M / 128);
        if (transB)
            gemm_f32_wmma<1><<<grid, 256, 0, stream>>>(A, lda, B, ldb, C, ldc,
                                                       aux, ldaux, tv, K, mode, alpha);
        else
            gemm_f32_wmma<0><<<grid, 256, 0, stream>>>(A, lda, B, ldb, C, ldc,
                                                       aux, ldaux, tv, K, mode, alpha);
    };

    // 1) LayerNorm (+1/sqrt(DHEAD) fold)
    layernorm_kernel<<<n, 256, 0, stream>>>(x, lnw, lnb, h, dim);

    // 2) Q/K projections:  q = h * wq^T,  k = h * wk^T   (NT GEMMs)
    gemm(h, dim, wq, dim, q, dim, n, dim, dim, 1, 0, 1.0f, nullptr, 0, nullptr);
    gemm(h, dim, wk, dim, k, dim, n, dim, dim, 1, 0, 1.0f, nullptr, 0, nullptr);

    // 3) Rotary embedding on q and k
    {
        int pairs = n * dim / 2;
        rope_kernel<<<(pairs + 255) / 256, 256, 0, stream>>>(q, sn, cs, n, dim);
        rope_kernel<<<(pairs + 255) / 256, 256, 0, stream>>>(k, sn, cs, n, dim);
    }

    // 4) per-head solver + gradient
    for (int hh = 0; hh < NHEAD; ++hh) {
        const float* hhp = h + hh * DHEAD;   // h_head: (n x 64), ld = dim

        // attention scores S = (q_h k_h^T)/8 -> causal softmax -> J (/256)
        gemm(q + hh * DHEAD, dim, k + hh * DHEAD, dim, J, n,
             n, n, DHEAD, 1, 0, 0.125f, nullptr, 0, nullptr);
        softmax_causal<<<n, 256, 0, stream>>>(J, n, 1.0f / 256.0f);

        set_ones<<<4, 256, 0, stream>>>(t, n);

        // 20 fixed-point steps + 1 final inverse at t_star
        for (int it = 0; it < 21; ++it) {
            // X1 = D^-1 + D^-1 J D^-1
            build_x1<<<n * n / 256, 256, 0, stream>>>(X, J, t, n);
            // Newton-Schulz step 1:  Y = V X ; X2 = 2X - X Y
            gemm(J, n, X,  n, Y,  n, n, n, n, 0, 1, 1.0f, X,  n, t);
            gemm(X, n, Y,  n, X2, n, n, n, n, 0, 2, 1.0f, X,  n, nullptr);
            // Newton-Schulz step 2:  Y = V X2 ; X = 2X2 - X2 Y
            gemm(J,  n, X2, n, Y, n, n, n, n, 0, 1, 1.0f, X2, n, t);
            gemm(X2, n, Y,  n, X, n, n, n, n, 0, 2, 1.0f, X2, n, nullptr);

            if (it < 20) {
                // W = Vinv * h_head ; t update
                gemm(X, n, hhp, dim, W, DHEAD, n, DHEAD, n,
                     0, 0, 1.0f, nullptr, 0, nullptr);
                update_t<<<4, 256, 0, stream>>>(t, X, W, n);
            }
        }

        // out_head = -0.25 (Vinv + Vinv^T) h_head
        symneg<<<n * n / 256, 256, 0, stream>>>(Y, X, n);
        gemm(Y, n, hhp, dim, out + hh * DHEAD, dim, n, DHEAD, n,
             0, 0, 1.0f, nullptr, 0, nullptr);
    }
}